// SOPTD_66133906423887
// MI455X (gfx1250) — compile-verified
//
#include <hip/hip_runtime.h>
#include <math.h>

typedef __attribute__((ext_vector_type(2))) float v2f;
typedef __attribute__((ext_vector_type(8))) float v8f;

__device__ __forceinline__ float wave_bcast(float v, int srcLane) {
    return __shfl(v, srcLane, 32);
}

__device__ __forceinline__ float wave_sum32(float v) {
    v += __shfl_xor(v, 16, 32);
    v += __shfl_xor(v, 8, 32);
    v += __shfl_xor(v, 4, 32);
    v += __shfl_xor(v, 2, 32);
    v += __shfl_xor(v, 1, 32);
    return v;
}

// out[n] = bias[n] + sum_i h[i] * W[i*32 + n], computed with V_WMMA_F32_16X16X4_F32.
// h_in is lane-distributed (lane i holds h[i]); return is lane-distributed too.
//
// Layout assumptions (CDNA5 ISA 7.12.2, 32-bit operands, wave32):
//   A 16x4 : VGPR j, lane l -> M = l%16, K = j + 2*(l>=16)
//   B 4x16 : VGPR j, lane l -> K = j + 2*(l>=16), N = l%16
//   C/D    : VGPR g, lanes 0-15 -> M=g, lanes 16-31 -> M=g+8, N = l%16
// A is the activation vector broadcast across all 16 rows, so every row of D
// equals the GEMV result; we read row M=0 (D vgpr0, lanes 0-15).
__device__ __forceinline__ float gemv32_f32_wmma(const float* __restrict__ W,
                                                 const float* __restrict__ bias,
                                                 float h_in) {
    const int lane = threadIdx.x;
    const int m  = lane & 15;
    const int hi = lane >> 4;          // 0 for lanes 0-15, 1 for lanes 16-31

    v8f c0 = {};   // N = 0..15
    v8f c1 = {};   // N = 16..31
    #pragma unroll
    for (int kc = 0; kc < 8; ++kc) {   // K chunks of 4
        v2f a;
        a.x = wave_bcast(h_in, 4 * kc + 0 + 2 * hi);
        a.y = wave_bcast(h_in, 4 * kc + 1 + 2 * hi);

        const int r0 = (4 * kc + 0 + 2 * hi) * 32;
        const int r1 = (4 * kc + 1 + 2 * hi) * 32;
        v2f b0, b1;
        b0.x = W[r0 + m];        b0.y = W[r1 + m];         // N half 0
        b1.x = W[r0 + m + 16];   b1.y = W[r1 + m + 16];    // N half 1

        c0 = __builtin_amdgcn_wmma_f32_16x16x4_f32(false, a, false, b0,
                                                   (short)0, c0, false, false);
        c1 = __builtin_amdgcn_wmma_f32_16x16x4_f32(false, a, false, b1,
                                                   (short)0, c1, false, false);
    }
    // Gather row M=0: out[0..15] lives in c0[0] lanes 0..15, out[16..31] in c1[0] lanes 0..15.
    float lo = wave_bcast(c0[0], m);
    float hv = wave_bcast(c1[0], m);
    return ((hi == 0) ? lo : hv) + bias[lane];
}

__global__ __launch_bounds__(32) void SOPTD_66133906423887_kernel(
    const float* __restrict__ states,
    const float* __restrict__ W1, const float* __restrict__ b1,
    const float* __restrict__ W2, const float* __restrict__ b2,
    const float* __restrict__ W3, const float* __restrict__ b3,
    const float* __restrict__ W4, const float* __restrict__ b4,
    const float* __restrict__ W5, const float* __restrict__ b5,
    const float* __restrict__ W6, const float* __restrict__ b6,
    const float* __restrict__ W7, const float* __restrict__ b7,
    const float* __restrict__ W8, const float* __restrict__ b8,
    const float* __restrict__ Kp1, const float* __restrict__ Kp2,
    const float* __restrict__ Tau1, const float* __restrict__ Tau2,
    const float* __restrict__ z1p, const float* __restrict__ z2p,
    float* __restrict__ out)
{
    const int lane = threadIdx.x;

    const float u   = states[0];
    const float x1  = states[1];
    const float x2  = states[2];
    const float x1p = states[3];
    const float x2p = states[4];

    // ---- MLP1 layer 1 (3 -> 4, linear), wave-uniform scalars ----
    float h4[4];
    #pragma unroll
    for (int j = 0; j < 4; ++j)
        h4[j] = b1[j] + u * W1[0 * 4 + j] + x1 * W1[1 * 4 + j] + x2 * W1[2 * 4 + j];

    // ---- MLP1 layer 2 (4 -> 32, tanh), lane n owns column n ----
    float h = b2[lane];
    #pragma unroll
    for (int k = 0; k < 4; ++k) h += h4[k] * W2[k * 32 + lane];
    h = tanhf(h);

    // ---- MLP1 layer 3 (32 -> 32, tanh) : WMMA f32 ----
    h = tanhf(gemv32_f32_wmma(W3, b3, h));

    // ---- MLP1 layer 4 (32 -> 2, linear) : butterfly reduce -> wave-uniform ----
    const float u60 = wave_sum32(h * W4[lane * 2 + 0]) + b4[0];
    const float u61 = wave_sum32(h * W4[lane * 2 + 1]) + b4[1];

    // ---- scalar second-order ODE (redundant per lane, all wave-uniform) ----
    const float kp1 = Kp1[0], kp2 = Kp2[0];
    const float t1 = Tau1[0], t2 = Tau2[0];
    const float zz1 = z1p[0], zz2 = z2p[0];
    const float x1pd = (kp1 / t1 * u60 - 2.0f * (zz1 * x1) - x1p / t1) / t1;
    const float x2pd = (kp2 / t2 * u61 - 2.0f * (zz2 * x2) - x2p / t2) / t2;
    const float xd[5] = { x1p, x2p, x1pd, x2pd, u };   // [x1_dot, x2_dot, x1_p_dot, x2_p_dot, u]

    // ---- MLP2 layer 1 (5 -> 32, linear) ----
    float g = b5[lane];
    #pragma unroll
    for (int k = 0; k < 5; ++k) g += xd[k] * W5[k * 32 + lane];

    // ---- MLP2 layers 2,3 (32 -> 32, tanh) : WMMA f32 ----
    g = tanhf(gemv32_f32_wmma(W6, b6, g));
    g = tanhf(gemv32_f32_wmma(W7, b7, g));

    // ---- MLP2 layer 4 (32 -> 4, linear) ----
    float g5[4];
    #pragma unroll
    for (int j = 0; j < 4; ++j)
        g5[j] = wave_sum32(g * W8[lane * 4 + j]) + b8[j];

    if (lane == 0) {
        out[0] = 0.0f;
        out[1] = g5[0];
        out[2] = g5[1];
        out[3] = g5[2];
        out[4] = g5[3];
    }
}

extern "C" void kernel_launch(void* const* d_in, const int* in_sizes, int n_in,
                              void* d_out, int out_size, void* d_ws, size_t ws_size,
                              hipStream_t stream) {
    (void)in_sizes; (void)n_in; (void)out_size; (void)d_ws; (void)ws_size;
    // dict order: t, states, W1,b1, W2,b2, W3,b3, W4,b4, W5,b5, W6,b6, W7,b7, W8,b8,
    //             Kp1, Kp2, Tau1, Tau2, z1, z2
    const float* states = (const float*)d_in[1];
    const float* W1 = (const float*)d_in[2];  const float* b1 = (const float*)d_in[3];
    const float* W2 = (const float*)d_in[4];  const float* b2 = (const float*)d_in[5];
    const float* W3 = (const float*)d_in[6];  const float* b3 = (const float*)d_in[7];
    const float* W4 = (const float*)d_in[8];  const float* b4 = (const float*)d_in[9];
    const float* W5 = (const float*)d_in[10]; const float* b5 = (const float*)d_in[11];
    const float* W6 = (const float*)d_in[12]; const float* b6 = (const float*)d_in[13];
    const float* W7 = (const float*)d_in[14]; const float* b7 = (const float*)d_in[15];
    const float* W8 = (const float*)d_in[16]; const float* b8 = (const float*)d_in[17];
    const float* Kp1 = (const float*)d_in[18]; const float* Kp2 = (const float*)d_in[19];
    const float* Tau1 = (const float*)d_in[20]; const float* Tau2 = (const float*)d_in[21];
    const float* z1 = (const float*)d_in[22];  const float* z2 = (const float*)d_in[23];

    SOPTD_66133906423887_kernel<<<1, 32, 0, stream>>>(
        states, W1, b1, W2, b2, W3, b3, W4, b4,
        W5, b5, W6, b6, W7, b7, W8, b8,
        Kp1, Kp2, Tau1, Tau2, z1, z2, (float*)d_out);
}